// DualPrompt_57011395887699
// MI455X (gfx1250) — compile-verified
//
#include <hip/hip_runtime.h>
#include <hip/hip_bf16.h>

typedef __attribute__((ext_vector_type(2))) float v2f;
typedef __attribute__((ext_vector_type(4))) float v4f;
typedef __attribute__((ext_vector_type(8))) float v8f;

// ---------------- problem constants ----------------
#define B_    128
#define D_    768
#define POOL_ 100
#define PPAD_ 112          // pool padded to 7*16 for WMMA tiling
#define LG_   6
#define LE_   6
#define GLEN_ 5
#define ELEN_ 20
#define H_    12
#define HD_   64
#define TOPK_ 5

#define GOUT_ELEMS (LG_ * 2 * B_ * H_ * GLEN_ * HD_)         // 5,898,240
#define EOUT_ELEMS (LE_ * 2 * B_ * H_ * (TOPK_ * ELEN_) * HD_) // 117,964,800

// ---------------- kernel 0: inverse L2 norms (wave per row) ----------------
__global__ void dp_norms_kernel(const float* __restrict__ q,
                                const float* __restrict__ keys,
                                float* __restrict__ rq,
                                float* __restrict__ rk) {
  int row  = blockIdx.x;     // 0..239
  int lane = threadIdx.x;    // 0..31 (wave32)
  const float* src;
  float* dst;
  if (row < B_) {
    src = q + row * D_;
    dst = rq + row;
  } else {
    int j = row - B_;
    if (j >= POOL_) { if (lane == 0) rk[j] = 0.0f; return; }
    src = keys + j * D_;
    dst = rk + j;
  }
  float s = 0.0f;
  #pragma unroll 4
  for (int k = lane; k < D_; k += 32) {
    float v = src[k];
    s = fmaf(v, v, s);
  }
  // wave32 butterfly reduction
  for (int m = 16; m >= 1; m >>= 1) s += __shfl_xor(s, m, 32);
  if (lane == 0) *dst = 1.0f / fmaxf(sqrtf(s), 1e-12f);
}

// ---------------- kernel 1: sims = qn @ kn^T via V_WMMA_F32_16X16X4_F32 ----
// grid (8, 7), one wave per 16x16 output tile, K=768 in steps of 4.
__global__ void dp_sims_wmma_kernel(const float* __restrict__ q,
                                    const float* __restrict__ keys,
                                    const float* __restrict__ rq,
                                    const float* __restrict__ rk,
                                    float* __restrict__ sims) {
  int lane = threadIdx.x;
  int half = lane >> 4;      // 0: K=k0+0..1, 1: K=k0+2..3
  int l16  = lane & 15;
  int m0 = blockIdx.x * 16;  // query rows
  int n0 = blockIdx.y * 16;  // key rows (= sims cols)

  int arow  = m0 + l16;
  int bcol  = n0 + l16;
  int bcolc = bcol < POOL_ ? bcol : (POOL_ - 1);  // clamp: padded cols never stored

  // 16x4 f32 A layout: lanes 0-15 rows M, vgpr0/1 = K+0/K+1; lanes 16-31 K+2/K+3
  const float* aptr = q    + arow  * D_ + 2 * half;
  const float* bptr = keys + bcolc * D_ + 2 * half;

  v8f acc = {};
  #pragma unroll 4
  for (int k0 = 0; k0 < D_; k0 += 4) {
    v2f a = *(const v2f*)(aptr + k0);
    v2f b = *(const v2f*)(bptr + k0);
    acc = __builtin_amdgcn_wmma_f32_16x16x4_f32(
        /*neg_a=*/false, a, /*neg_b=*/false, b,
        /*c_mod=*/(short)0, acc, /*reuse_a=*/false, /*reuse_b=*/false);
  }

  // D layout: vgpr r -> M = r + 8*half, N = l16 (+n0)
  if (bcol < POOL_) {
    float rkc = rk[bcol];
    #pragma unroll
    for (int r = 0; r < 8; ++r) {
      int row = m0 + r + 8 * half;
      sims[row * PPAD_ + bcol] = acc[r] * rq[row] * rkc;
    }
  }
}

// ---------------- kernel 2: top-5 per row (matches jax.lax.top_k ties) -----
__global__ void dp_topk_kernel(const float* __restrict__ sims,
                               int* __restrict__ idx) {
  int row = blockIdx.x * blockDim.x + threadIdx.x;
  if (row >= B_) return;
  const float* s = sims + row * PPAD_;
  int chosen[TOPK_];
  for (int k = 0; k < TOPK_; ++k) {
    float best = -3.402823466e38f;
    int bi = 0;
    for (int p = 0; p < POOL_; ++p) {
      bool taken = false;
      for (int j = 0; j < k; ++j) taken |= (chosen[j] == p);
      float v = s[p];
      // strict '>' while scanning ascending p => lowest index among ties
      if (!taken && v > best) { best = v; bi = p; }
    }
    chosen[k] = bi;
    idx[row * TOPK_ + k] = bi;
  }
}

// ---------------- kernel 3: g_out broadcast (float4, NT stores) ------------
// out[l,s,b,h,g,:] = g_prompt[l,s,g,h,:]
__global__ void dp_gout_kernel(const v4f* __restrict__ g, v4f* __restrict__ out) {
  int o4 = blockIdx.x * 256 + threadIdx.x;   // exact grid: GOUT_ELEMS/4 threads
  int d4 = o4 & 15;
  int t  = o4 >> 4;
  int gg = t % GLEN_; t /= GLEN_;
  int h  = t % H_;    t /= H_;
  t /= B_;                                   // drop b (broadcast)
  int s  = t & 1;
  int l  = t >> 1;
  v4f v = g[(((l * 2 + s) * GLEN_ + gg) * H_ + h) * 16 + d4];
  __builtin_nontemporal_store(v, out + o4);
}

// ---------------- kernel 4: e_out gather (float4, NT stores) ---------------
// out[l,s,b,h,k*elen+e,:] = pool[idx[b][k], l, s, e, h, :]
__global__ void dp_eout_kernel(const v4f* __restrict__ pool,
                               const int* __restrict__ idx,
                               v4f* __restrict__ out) {
  int o4 = blockIdx.x * 256 + threadIdx.x;   // exact grid: EOUT_ELEMS/4 threads
  int d4 = o4 & 15;
  int t  = o4 >> 4;
  int ke = t % (TOPK_ * ELEN_); t /= (TOPK_ * ELEN_);
  int h  = t % H_;              t /= H_;
  int b  = t & (B_ - 1);        t >>= 7;
  int s  = t & 1;
  int l  = t >> 1;
  int k  = ke / ELEN_;
  int e  = ke % ELEN_;
  int p  = idx[b * TOPK_ + k];
  v4f v = pool[((((p * LE_ + l) * 2 + s) * ELEN_ + e) * H_ + h) * 16 + d4];
  __builtin_nontemporal_store(v, out + o4);
}

extern "C" void kernel_launch(void* const* d_in, const int* in_sizes, int n_in,
                              void* d_out, int out_size, void* d_ws, size_t ws_size,
                              hipStream_t stream) {
  const float* q    = (const float*)d_in[0];   // [128, 768]
  const float* g    = (const float*)d_in[1];   // [6, 2, 5, 12, 64]
  const float* pool = (const float*)d_in[2];   // [100, 6, 2, 20, 12, 64]
  const float* keys = (const float*)d_in[3];   // [100, 768]

  float* out_g = (float*)d_out;                // [6,2,128,12,5,64]
  float* out_e = out_g + GOUT_ELEMS;           // [6,2,128,12,100,64]

  // workspace layout (floats): rq[128] | rk[112] | sims[128*112] | idx[640]
  float* rq   = (float*)d_ws;
  float* rk   = rq + B_;
  float* sims = rk + PPAD_;
  int*   idx  = (int*)(sims + B_ * PPAD_);

  // 1) inverse norms: one wave per row (128 query + 112 key slots)
  dp_norms_kernel<<<dim3(B_ + PPAD_), dim3(32), 0, stream>>>(q, keys, rq, rk);

  // 2) cosine sims via f32 WMMA: 8 x 7 tiles, one wave each
  dp_sims_wmma_kernel<<<dim3(B_ / 16, PPAD_ / 16), dim3(32), 0, stream>>>(
      q, keys, rq, rk, sims);

  // 3) top-5 indices per row
  dp_topk_kernel<<<dim3(1), dim3(B_), 0, stream>>>(sims, idx);

  // 4) g_out broadcast copy (~24 MB, NT stores)
  dp_gout_kernel<<<dim3(GOUT_ELEMS / 4 / 256), dim3(256), 0, stream>>>(
      (const v4f*)g, (v4f*)out_g);

  // 5) e_out gather copy (~472 MB write; pool stays L2-resident via NT stores)
  dp_eout_kernel<<<dim3(EOUT_ELEMS / 4 / 256), dim3(256), 0, stream>>>(
      (const v4f*)pool, idx, (v4f*)out_e);
}